// DWTFM_6700148982261
// MI455X (gfx1250) — compile-verified
//
#include <hip/hip_runtime.h>
#include <hip/hip_bf16.h>
#include <stdint.h>

// Fused Haar DWT->IDWT:
//   out[i][j] = x1[i][j] + 0.25 * (sum2x2(x0 block) - sum2x2(x1 block))
// Streaming kernel at the HBM roofline (~151 MB moved, ~6.5us @ 23.3 TB/s).
// Data movement uses the CDNA5 async-to-LDS engine (ASYNCcnt) with b128 lines.

#define IMG_W 512
#define IMG_H 512
#define BLOCK_THREADS 256

__global__ __launch_bounds__(BLOCK_THREADS) void haar_fuse_kernel(
    const float* __restrict__ x0, const float* __restrict__ x1,
    float* __restrict__ out, unsigned nTiles)
{
    // Four staging planes (x0-top, x0-bot, x1-top, x1-bot), each 256 x float4.
    // Lane stride inside a plane = 16B -> conflict-free ds_read_b128 phases.
    __shared__ float4 sm[4][BLOCK_THREADS];

    const unsigned tid = threadIdx.x;
    const unsigned t   = blockIdx.x * BLOCK_THREADS + tid;
    if (t >= nTiles) return;

    // Tile = 2 rows x 4 cols (two 2x2 Haar blocks). All pow-2 -> pure shifts.
    const unsigned img = t >> 15;          // / (256 row-pairs * 128 tile-cols)
    const unsigned rem = t & 32767u;
    const unsigned tr  = rem >> 7;         // / 128 tile-cols
    const unsigned tc  = rem & 127u;

    const unsigned base = (img << 18) + (tr << 10) + (tc << 2);

    const float* a_t = x0 + base;          // x0, top row of block pair
    const float* a_b = a_t + IMG_W;        // x0, bottom row
    const float* b_t = x1 + base;          // x1, top row
    const float* b_b = b_t + IMG_W;        // x1, bottom row

    // LDS byte addresses of this lane's slots (low 32 bits of the flat LDS
    // pointer are the LDS offset on gfx1250).
    const unsigned l0 = (unsigned)(uintptr_t)(&sm[0][0]) + tid * 16u;

    // CDNA5 async data mover: global -> LDS, 128b per lane, tracked by ASYNCcnt.
    asm volatile(
        "global_load_async_to_lds_b128 %0, %4, off\n\t"
        "global_load_async_to_lds_b128 %1, %5, off\n\t"
        "global_load_async_to_lds_b128 %2, %6, off\n\t"
        "global_load_async_to_lds_b128 %3, %7, off"
        :
        : "v"(l0), "v"(l0 + 4096u), "v"(l0 + 8192u), "v"(l0 + 12288u),
          "v"(a_t), "v"(a_b), "v"(b_t), "v"(b_b)
        : "memory");
    asm volatile("s_wait_asynccnt 0" ::: "memory");

    const float4 T0 = sm[0][tid];          // x0 top
    const float4 B0 = sm[1][tid];          // x0 bottom
    const float4 T1 = sm[2][tid];          // x1 top
    const float4 B1 = sm[3][tid];          // x1 bottom

    // Per-2x2-block correction: 0.25 * (sumA - sumB)
    const float sL = 0.25f * ((T0.x + T0.y + B0.x + B0.y) -
                              (T1.x + T1.y + B1.x + B1.y));
    const float sR = 0.25f * ((T0.z + T0.w + B0.z + B0.w) -
                              (T1.z + T1.w + B1.z + B1.w));

    const float4 ot = make_float4(T1.x + sL, T1.y + sL, T1.z + sR, T1.w + sR);
    const float4 ob = make_float4(B1.x + sL, B1.y + sL, B1.z + sR, B1.w + sR);

    *(float4*)(out + base)         = ot;
    *(float4*)(out + base + IMG_W) = ob;
}

extern "C" void kernel_launch(void* const* d_in, const int* in_sizes, int n_in,
                              void* d_out, int out_size, void* d_ws, size_t ws_size,
                              hipStream_t stream) {
    (void)in_sizes; (void)n_in; (void)d_ws; (void)ws_size;
    const float* x0 = (const float*)d_in[0];
    const float* x1 = (const float*)d_in[1];
    float* out = (float*)d_out;

    const unsigned nTiles = (unsigned)(out_size / 8);   // 8 outputs per thread
    const unsigned blocks = (nTiles + BLOCK_THREADS - 1) / BLOCK_THREADS;
    haar_fuse_kernel<<<blocks, BLOCK_THREADS, 0, stream>>>(x0, x1, out, nTiles);
}